// CustomGCNLayer_13692355740141
// MI455X (gfx1250) — compile-verified
//
#include <hip/hip_runtime.h>

#define N_NODES 50000
#define N_EDGES 1600000
#define D       128
#define M_TILES ((N_NODES + 15) / 16)   // 3125 exactly

typedef __attribute__((ext_vector_type(2))) float v2f;
typedef __attribute__((ext_vector_type(8))) float v8f;

// ---------------------------------------------------------------------------
// GEMM: h[n,o] = sum_k x[n,k] * W[o,k] + b[o]   (x:[N,128] rm, W:[128,128] rm)
// One wave per 16-row M tile; 8 accumulator tiles cover all 128 output cols.
// Uses V_WMMA_F32_16X16X4_F32 (exact fp32) — 256 WMMAs per wave.
//
// f32 A (16x4) layout (ISA 7.12.2): lanes 0-15 -> K={k0,k0+1} in VGPR0/1,
// lanes 16-31 -> K={k0+2,k0+3}. B (4x16) mirrored: lane = N across lanes,
// VGPR v holds K = half*2 + v. C/D 16x16: VGPR r, lanes 0-15: M=r, N=lane;
// lanes 16-31: M=r+8, N=lane-16.
// ---------------------------------------------------------------------------
__global__ __launch_bounds__(256) void gcn_gemm_wmma_f32(
    const float* __restrict__ x, const float* __restrict__ W,
    const float* __restrict__ b, float* __restrict__ h) {
  const int lane = threadIdx.x & 31;
  const int wave = threadIdx.x >> 5;
  const int mtile = blockIdx.x * 8 + wave;
  if (mtile >= M_TILES) return;               // whole-wave exit: EXEC stays all-1s
  const int m0 = mtile * 16;

  const int half = lane >> 4;                 // which K pair this lane holds
  const int sub  = lane & 15;                 // M index (A) / N index (B, C)

  v8f acc[8] = {};

  const float* xrow = x + (size_t)(m0 + sub) * D;
  for (int k0 = 0; k0 < D; k0 += 4) {
    const int ka = k0 + half * 2;
    v2f a;
    a.x = xrow[ka + 0];                       // contiguous pair -> b64 load
    a.y = xrow[ka + 1];
#pragma unroll
    for (int t = 0; t < 8; ++t) {
      const int o = t * 16 + sub;             // output column for this lane
      v2f bb;
      bb.x = W[(size_t)o * D + ka + 0];       // B[k][o] = W[o][k]
      bb.y = W[(size_t)o * D + ka + 1];
      acc[t] = __builtin_amdgcn_wmma_f32_16x16x4_f32(
          /*neg_a=*/false, a, /*neg_b=*/false, bb,
          /*c_mod=*/(short)0, acc[t], /*reuse_a=*/false, /*reuse_b=*/false);
    }
  }

#pragma unroll
  for (int t = 0; t < 8; ++t) {
    const int o = t * 16 + sub;
    const float bias = b[o];
#pragma unroll
    for (int r = 0; r < 8; ++r) {
      const int m = r + half * 8;
      h[(size_t)(m0 + m) * D + o] = acc[t][r] + bias;
    }
  }
}

// ---------------------------------------------------------------------------
// Scatter-add: one wave per edge, each lane owns 4 contiguous feature dims.
// h[src] gathered as float4 (L2-resident), summed into out[dst] with HW f32
// atomics (global_atomic_add_f32). Lane 0 bumps the neighbor count.
// ---------------------------------------------------------------------------
__global__ __launch_bounds__(256) void gcn_scatter(
    const long long* __restrict__ ei, const float* __restrict__ h,
    float* __restrict__ sums, float* __restrict__ counts) {
  const int lane = threadIdx.x & 31;
  const int e = blockIdx.x * 8 + (threadIdx.x >> 5);
  if (e >= N_EDGES) return;
  const int src = (int)ei[e];                 // row = source node
  const int dst = (int)ei[(size_t)N_EDGES + e];  // col = destination node

  const float4 v = ((const float4*)(h + (size_t)src * D))[lane];
  float* o = sums + (size_t)dst * D + lane * 4;
  unsafeAtomicAdd(o + 0, v.x);
  unsafeAtomicAdd(o + 1, v.y);
  unsafeAtomicAdd(o + 2, v.z);
  unsafeAtomicAdd(o + 3, v.w);
  if (lane == 0) unsafeAtomicAdd(counts + dst, 1.0f);
}

// ---------------------------------------------------------------------------
// Finalize: out = sums/count if count>0 else h  (isolated nodes keep own h).
// ---------------------------------------------------------------------------
__global__ __launch_bounds__(256) void gcn_finalize(
    const float* __restrict__ h, const float* __restrict__ counts,
    float* __restrict__ out) {
  const int i = blockIdx.x * blockDim.x + threadIdx.x;
  if (i >= N_NODES * D) return;
  const int node = i >> 7;                    // D == 128
  const float c = counts[node];
  const float s = out[i];
  out[i] = (c > 0.0f) ? (s / c) : h[i];
}

extern "C" void kernel_launch(void* const* d_in, const int* in_sizes, int n_in,
                              void* d_out, int out_size, void* d_ws, size_t ws_size,
                              hipStream_t stream) {
  const float*     x  = (const float*)d_in[0];      // [N, 128] f32
  const long long* ei = (const long long*)d_in[1];  // [2, E] i64
  const float*     W  = (const float*)d_in[2];      // [128, 128] f32
  const float*     b  = (const float*)d_in[3];      // [128] f32
  float* out = (float*)d_out;                       // [N, 128] f32

  float* h      = (float*)d_ws;                     // N*D floats
  float* counts = h + (size_t)N_NODES * D;          // N floats

  // Deterministic per-call init (accumulate sums directly into d_out).
  hipMemsetAsync(out, 0, sizeof(float) * (size_t)N_NODES * D, stream);
  hipMemsetAsync(counts, 0, sizeof(float) * (size_t)N_NODES, stream);

  gcn_gemm_wmma_f32<<<(M_TILES + 7) / 8, 256, 0, stream>>>(x, W, b, h);
  gcn_scatter<<<(N_EDGES + 7) / 8, 256, 0, stream>>>(ei, h, out, counts);
  gcn_finalize<<<((size_t)N_NODES * D + 255) / 256, 256, 0, stream>>>(h, counts, out);
}